// HarmonicDecomposition_46145128628320
// MI455X (gfx1250) — compile-verified
//
#include <hip/hip_runtime.h>
#include <math.h>

// Problem dims (from reference)
#define BATCH   8
#define BOX     160
#define KTOT    (BOX * BOX)      // 25600  (pixel reduction dim)
#define NH      20
#define LH      20
#define NLR     (NH * LH * 2)    // 800    (mode dim)
#define ROW_TILES (NLR / 16)     // 50
#define K_CHUNKS  16
#define K_CHUNK   (KTOT / K_CHUNKS) // 1600
#define COL_TILES (KTOT / 16)    // 1600

typedef __attribute__((ext_vector_type(2))) float v2f;
typedef __attribute__((ext_vector_type(8))) float v8f;

// ---------------------------------------------------------------------------
// Kernel 0: zero-pad x (8 x K) -> xpad (16 x K) so the WMMA B-operand load
// in the projection loop is unconditional (no exec-mask divergence).
// ---------------------------------------------------------------------------
__global__ void __launch_bounds__(256)
hd_padx_kernel(const float* __restrict__ x, float* __restrict__ xpad) {
    const int idx = (int)(blockIdx.x * blockDim.x + threadIdx.x);
    if (idx >= 16 * KTOT) return;
    const int r = idx / KTOT;
    const int k = idx - r * KTOT;
    xpad[idx] = (r < BATCH) ? x[(size_t)r * KTOT + k] : 0.f;
}

// ---------------------------------------------------------------------------
// Kernel 1: coeffs^T[nlr, b] = sum_k basis[nlr, k] * xpad[b, k]
// A = basis rows (M=nlr, K=pixels), B = xpad^T (K=pixels, N=16, cols 8..15 = 0)
// One wave per (row_tile, k_chunk); partial 16x16 f32 accumulators -> ws.
// ---------------------------------------------------------------------------
__global__ void __launch_bounds__(256)
hd_proj_kernel(const float* __restrict__ xpad, const float* __restrict__ basis,
               float* __restrict__ partials) {
    const int wave = (int)((blockIdx.x * blockDim.x + threadIdx.x) >> 5);
    const int lane = (int)(threadIdx.x & 31u);
    const int tile  = wave / K_CHUNKS;   // 0..49  (nlr row tile)
    const int chunk = wave % K_CHUNKS;   // 0..15  (K chunk)

    const int mrow  = (lane & 15);            // M within tile (both half-waves)
    const int khalf = (lane >> 4) * 2;        // lanes 0-15: K+{0,1}; 16-31: K+{2,3}

    // A: lane streams basis row (tile*16 + mrow), contiguous float2 every 4 K
    const float* aPtr = basis + (size_t)(tile * 16 + mrow) * KTOT
                              + (size_t)chunk * K_CHUNK + khalf;
    // B: lane holds xpad[b = lane&15][k] (zeros baked in for b >= 8)
    const float* bPtr = xpad + (size_t)(lane & 15) * KTOT
                             + (size_t)chunk * K_CHUNK + khalf;

    v8f acc = {};
    for (int ko = 0; ko < K_CHUNK; ko += 16) {
        __builtin_prefetch(aPtr + 1024, 0, 3);   // ~4 KB ahead on the stream
#pragma unroll
        for (int ki = 0; ki < 16; ki += 4) {
            v2f a = *(const v2f*)(aPtr + ki);    // A[m][khalf], A[m][khalf+1]
            v2f b = *(const v2f*)(bPtr + ki);    // B[khalf][n], B[khalf+1][n]
            acc = __builtin_amdgcn_wmma_f32_16x16x4_f32(
                      false, a, false, b, (short)0, acc, false, false);
        }
        aPtr += 16;
        bPtr += 16;
    }

    // D layout: vgpr v -> M = v + 8*(lane>=16), N = lane&15
    float* p = partials + (size_t)(tile * K_CHUNKS + chunk) * 256;
    const int n = lane & 15;
    const int mbase = (lane >> 4) * 8;
#pragma unroll
    for (int v = 0; v < 8; ++v) p[(mbase + v) * 16 + n] = acc[v];
}

// ---------------------------------------------------------------------------
// Kernel 2: sum K-chunk partials -> coeffs16[nlr][16] (cols 8..15 stay zero
// automatically since B-cols 8..15 were zero in the projection).
// ---------------------------------------------------------------------------
__global__ void __launch_bounds__(256)
hd_reduce_kernel(const float* __restrict__ partials, float* __restrict__ coeffs16) {
    const int idx = (int)(blockIdx.x * blockDim.x + threadIdx.x);
    if (idx >= ROW_TILES * 256) return;
    const int tile = idx >> 8;
    const int mn   = idx & 255;            // m*16 + n
    float s = 0.f;
#pragma unroll
    for (int c = 0; c < K_CHUNKS; ++c)
        s += partials[(size_t)(tile * K_CHUNKS + c) * 256 + mn];
    coeffs16[(size_t)tile * 256 + mn] = s; // [nlr][16] padded layout
}

// ---------------------------------------------------------------------------
// Kernel 3: complex rotation per (n, l, b); in-place on coeffs16[nlr][16],
// touches only cols b < 8 (pad columns remain zero).
// ---------------------------------------------------------------------------
__global__ void __launch_bounds__(256)
hd_rotate_kernel(float* __restrict__ coeffs16, const float* __restrict__ angles) {
    const int idx = (int)(blockIdx.x * blockDim.x + threadIdx.x);
    if (idx >= NH * LH * BATCH) return;
    const int b = idx & (BATCH - 1);
    const int l = (idx >> 3) % LH;
    const int n = idx / (LH * BATCH);
    float sa, ca;
    sincosf(angles[b] * (float)l, &sa, &ca);
    const int base = ((n * LH + l) * 2) * 16 + b;  // r=0 slot; r=1 at +16
    const float c0 = coeffs16[base];
    const float c1 = coeffs16[base + 16];
    coeffs16[base]      = c0 * ca + c1 * sa;
    coeffs16[base + 16] = c1 * ca - c0 * sa;
}

// ---------------------------------------------------------------------------
// Kernel 4: out[b][ij] = sum_nlr rot[nlr][b] * basis[nlr][ij]
// A = rot^T from padded coeffs16 (M=16, rows 8..15 zero), B = basis
// (row-major K x N) -> fully coalesced, L2-resident on this second pass.
// One wave per 16-column tile of ij; full K=800 loop; direct store.
// ---------------------------------------------------------------------------
__global__ void __launch_bounds__(256)
hd_recon_kernel(const float* __restrict__ rot16, const float* __restrict__ basis,
                float* __restrict__ out) {
    const int wave = (int)((blockIdx.x * blockDim.x + threadIdx.x) >> 5);
    const int lane = (int)(threadIdx.x & 31u);
    const int col0  = wave * 16;              // ij column tile
    const int khalf = (lane >> 4) * 2;
    const int nIdx  = lane & 15;

    const float* bBase = basis + (size_t)khalf * KTOT + col0 + nIdx;
    const float* aBase = rot16 + (size_t)khalf * 16 + (lane & 15);

    v8f acc = {};
    for (int ko = 0; ko < NLR; ko += 16) {
        __builtin_prefetch(bBase + (size_t)(ko + 16) * KTOT, 0, 3);
#pragma unroll
        for (int ki = 0; ki < 16; ki += 4) {
            const int k = ko + ki;
            v2f a;
            a.x = aBase[(size_t)k * 16];          // rot[k+khalf][m]
            a.y = aBase[(size_t)(k + 1) * 16];    // rot[k+khalf+1][m]
            v2f b;
            b.x = bBase[(size_t)k * KTOT];        // basis[k+khalf][col]
            b.y = bBase[(size_t)(k + 1) * KTOT];  // basis[k+khalf+1][col]
            acc = __builtin_amdgcn_wmma_f32_16x16x4_f32(
                      false, a, false, b, (short)0, acc, false, false);
        }
    }

    // Only lanes 0-15 carry valid M (=0..7); vgpr v -> batch b = v
    if (lane < 16) {
#pragma unroll
        for (int v = 0; v < 8; ++v)
            out[(size_t)v * KTOT + col0 + nIdx] = acc[v];
    }
}

// ---------------------------------------------------------------------------
extern "C" void kernel_launch(void* const* d_in, const int* in_sizes, int n_in,
                              void* d_out, int out_size, void* d_ws, size_t ws_size,
                              hipStream_t stream) {
    const float* x      = (const float*)d_in[0];  // [8][1][160][160]
    const float* basis  = (const float*)d_in[1];  // [20][20][2][160][160]
    const float* angles = (const float*)d_in[2];  // [8]
    float* out = (float*)d_out;                   // [8][1][160][160]

    // Workspace layout (floats):
    //   xpad     : 16 * 25600            = 409600
    //   partials : 50 * 16 * 256         = 204800
    //   coeffs16 : 800 * 16              =  12800
    float* xpad     = (float*)d_ws;
    float* partials = xpad + (size_t)16 * KTOT;
    float* coeffs16 = partials + (size_t)ROW_TILES * K_CHUNKS * 256;

    // 0) zero-pad x rows 8..15
    hd_padx_kernel<<<(16 * KTOT + 255) / 256, 256, 0, stream>>>(x, xpad);

    // 1) projection: 50 row tiles x 16 K-chunks = 800 waves = 100 blocks x 256
    hd_proj_kernel<<<(ROW_TILES * K_CHUNKS) / 8, 256, 0, stream>>>(xpad, basis, partials);

    // 2) reduce partials -> coeffs16 : 50*256 threads
    hd_reduce_kernel<<<ROW_TILES, 256, 0, stream>>>(partials, coeffs16);

    // 3) rotate modes in place : 3200 threads
    hd_rotate_kernel<<<(NH * LH * BATCH + 255) / 256, 256, 0, stream>>>(coeffs16, angles);

    // 4) reconstruction: 1600 column tiles, 1 wave each = 200 blocks x 256
    hd_recon_kernel<<<COL_TILES / 8, 256, 0, stream>>>(coeffs16, basis, out);
}